// DurationAwareFocalLoss_80341658239296
// MI455X (gfx1250) — compile-verified
//
#include <hip/hip_runtime.h>

// DurationAwareFocalLoss on MI455X (gfx1250, wave32).
// Memory-bound streaming reduction: 268 MB in -> 1 float out.
// Roofline: 268 MB / 23.3 TB/s ~= 11.5 us floor; compute ~0.5 GFLOP (trivial).
// Strategy: b128 non-temporal loads, 32-bit indexing, uniform (SALU) loop
// bounds, minimal VALU/elem, WMMA-assisted wave reduction, deterministic
// two-pass (block partials in d_ws, final single-block kernel).

typedef __attribute__((ext_vector_type(4))) float v4f;
typedef __attribute__((ext_vector_type(2))) float v2f;
typedef __attribute__((ext_vector_type(8))) float v8f;

// ALPHA=0.25 GAMMA=2 DELTA=0.5, weights LUT {1,4,3,2}
// focal_w * base  ==  d<0.5 ? 4*d^4 : d-0.25
// weight          ==  (rint(t) < 0.5) ? 1 : 5 - rint(t)
__device__ __forceinline__ float elem_loss(float p, float t) {
    float d  = __builtin_fabsf(p - t);
    float d2 = d * d;
    float quart = 4.0f * (d2 * d2);
    float lin   = d - 0.25f;
    float fb = (d < 0.5f) ? quart : lin;
    float c = __builtin_rintf(t);           // round-half-even == jnp.round
    c = fminf(fmaxf(c, 0.0f), 3.0f);
    float w = (c < 0.5f) ? 1.0f : (5.0f - c);
    return fb * w;
}

// Full-wave (32 lane) sum via V_WMMA_F32_16X16X4_F32.
// A(16x4): lanes0-15 supply K=0 (a[0]) / K=1 (a[1]); lanes16-31 K=2/K=3.
// With a = {s, 0}:  A[m][0]=s_m, A[m][2]=s_{m+16}.  B = ones(4x16) gives
// D[m][n] = s_m + s_{m+16} for all n.  D layout: lane L, vgpr r holds
// M = r (+8 for upper half) -> summing d[0..7] in-lane yields the half-sums;
// one shfl_xor(16) completes the 32-lane total in every lane.
// EXEC must be all ones here (callers guarantee full blocks, reconverged).
__device__ __forceinline__ float wave_reduce_wmma(float s) {
    v2f a; a[0] = s;    a[1] = 0.0f;
    v2f b; b[0] = 1.0f; b[1] = 1.0f;
    v8f c = {};
    v8f d = __builtin_amdgcn_wmma_f32_16x16x4_f32(
        /*neg_a=*/false, a, /*neg_b=*/false, b,
        /*c_mod=*/(short)0, c, /*reuse_a=*/false, /*reuse_b=*/false);
    float u = ((d[0] + d[1]) + (d[2] + d[3])) + ((d[4] + d[5]) + (d[6] + d[7]));
    u += __shfl_xor(u, 16, 32);
    return u;
}

__device__ __forceinline__ void block_reduce_store(float s, float* dst, float scale) {
    __shared__ float red[8];                 // 256 threads = 8 wave32
    float ws = wave_reduce_wmma(s);
    int wave = threadIdx.x >> 5;
    int lane = threadIdx.x & 31;
    if (lane == 0) red[wave] = ws;
    __syncthreads();
    if (threadIdx.x == 0) {
        float tot = 0.0f;
        #pragma unroll
        for (int i = 0; i < 8; ++i) tot += red[i];
        *dst = tot * scale;
    }
}

__global__ void __launch_bounds__(256)
dafl_partial_kernel(const float* __restrict__ p, const float* __restrict__ t,
                    float* __restrict__ partial, int n) {
    const int n4     = n >> 2;                                  // # float4 pairs
    const int tid    = (int)(blockIdx.x * blockDim.x + threadIdx.x);
    const int stride = (int)(gridDim.x * blockDim.x);
    const int iters  = n4 / stride;                             // uniform -> SALU loop
    const int rem    = n4 - iters * stride;                     // uniform

    const v4f* __restrict__ p4 = (const v4f*)p;
    const v4f* __restrict__ t4 = (const v4f*)t;

    float s = 0.0f;
    int idx = tid;
    int k = 0;
    // 4-deep unroll: 8 b128 NT loads in flight per wave per iteration.
    for (; k + 4 <= iters; k += 4, idx += 4 * stride) {
        #pragma unroll
        for (int u = 0; u < 4; ++u) {
            int j = idx + u * stride;
            v4f pv = __builtin_nontemporal_load(p4 + j);
            v4f tv = __builtin_nontemporal_load(t4 + j);
            s += elem_loss(pv[0], tv[0]);
            s += elem_loss(pv[1], tv[1]);
            s += elem_loss(pv[2], tv[2]);
            s += elem_loss(pv[3], tv[3]);
        }
    }
    for (; k < iters; ++k, idx += stride) {
        v4f pv = __builtin_nontemporal_load(p4 + idx);
        v4f tv = __builtin_nontemporal_load(t4 + idx);
        s += elem_loss(pv[0], tv[0]);
        s += elem_loss(pv[1], tv[1]);
        s += elem_loss(pv[2], tv[2]);
        s += elem_loss(pv[3], tv[3]);
    }
    // remainder float4s (n4 % stride of them): single divergent-guard load.
    if (tid < rem) {
        v4f pv = __builtin_nontemporal_load(p4 + idx);
        v4f tv = __builtin_nontemporal_load(t4 + idx);
        s += elem_loss(pv[0], tv[0]);
        s += elem_loss(pv[1], tv[1]);
        s += elem_loss(pv[2], tv[2]);
        s += elem_loss(pv[3], tv[3]);
    }
    // scalar tail (n not multiple of 4); reconverges before the WMMA.
    for (int e = (n4 << 2) + tid; e < n; e += stride) {
        s += elem_loss(p[e], t[e]);
    }

    block_reduce_store(s, partial + blockIdx.x, 1.0f);
}

__global__ void __launch_bounds__(256)
dafl_final_kernel(const float* __restrict__ partial, int nparts,
                  float* __restrict__ out, float scale) {
    float s = 0.0f;
    for (int i = (int)threadIdx.x; i < nparts; i += (int)blockDim.x) s += partial[i];
    block_reduce_store(s, out, scale);
}

extern "C" void kernel_launch(void* const* d_in, const int* in_sizes, int n_in,
                              void* d_out, int out_size, void* d_ws, size_t ws_size,
                              hipStream_t stream) {
    const float* y_pred = (const float*)d_in[0];
    const float* y_true = (const float*)d_in[1];
    float* out = (float*)d_out;
    float* ws  = (float*)d_ws;
    const int n = in_sizes[0];                    // B*N flattened (2^25)

    int blocks = 2048;                            // 2048 blocks x 8 wave32: fills the WGPs
    size_t need = (size_t)blocks * sizeof(float);
    if (need > ws_size) blocks = (int)(ws_size / sizeof(float));
    if (blocks < 1) blocks = 1;

    dafl_partial_kernel<<<blocks, 256, 0, stream>>>(y_pred, y_true, ws, n);
    // mean * ALPHA folded into one scale
    dafl_final_kernel<<<1, 256, 0, stream>>>(ws, blocks, out, 0.25f / (float)n);
}